// SAGE_24232205484235
// MI455X (gfx1250) — compile-verified
//
#include <hip/hip_runtime.h>

#define N_NODES 50000
#define N_EDGES 800000
#define FEATS   128     // input dim of every layer (128 -> 128 -> 128 -> 64)

typedef __attribute__((ext_vector_type(2))) float v2f;
typedef __attribute__((ext_vector_type(8))) float v8f;

// ---------------------------------------------------------------- utilities
__global__ void sage_fill_zero(float* __restrict__ p, int n) {
  int i = blockIdx.x * blockDim.x + threadIdx.x;
  if (i < n) p[i] = 0.0f;
}

__global__ void sage_degree(const int* __restrict__ dst, float* __restrict__ deg) {
  int e = blockIdx.x * blockDim.x + threadIdx.x;
  if (e < N_EDGES) atomicAdd(&deg[dst[e]], 1.0f);
}

__global__ void sage_invdeg(float* __restrict__ deg) {
  int i = blockIdx.x * blockDim.x + threadIdx.x;
  if (i < N_NODES) deg[i] = 1.0f / fmaxf(deg[i], 1.0f);
}

// ------------------------------------------------- edge-parallel aggregation
// One wave32 per edge; each lane moves 4 contiguous floats (32*4 = 128 feats).
// h and agg are L2-resident (25.6 MB each vs 192 MB L2), so both the float4
// gather and the scatter atomics are L2 traffic, not HBM.
__global__ void sage_aggregate(const float* __restrict__ h,
                               const int* __restrict__ src,
                               const int* __restrict__ dst,
                               float* __restrict__ agg) {
  int tid  = blockIdx.x * blockDim.x + threadIdx.x;
  int e    = tid >> 5;
  int lane = tid & 31;
  if (e >= N_EDGES) return;
  int s = src[e];
  int d = dst[e];
  const float4 v = reinterpret_cast<const float4*>(h + (size_t)s * FEATS)[lane];
  float* ap = agg + (size_t)d * FEATS + lane * 4;
  atomicAdd(ap + 0, v.x);
  atomicAdd(ap + 1, v.y);
  atomicAdd(ap + 2, v.z);
  atomicAdd(ap + 3, v.w);
}

// ------------------------------------------------------ WMMA fused dual-GEMM
// out[M,N] = relu?( h[M,:]@Wself + (agg[M,:]*invdeg[M])@Wneigh + b[N] )
// One wave per 16x16 output tile, accumulating K in steps of 4 with
// V_WMMA_F32_16X16X4_F32 (full fp32 precision, matches the reference).
//
// A-frag 16x4 f32 (2 VGPR): lanes 0-15 row=lane, K={0,1}; lanes 16-31 K={2,3}
// B-frag  4x16 f32 (2 VGPR): lanes 0-15 col=lane, K={0,1}; lanes 16-31 K={2,3}
// C/D 16x16 f32 (8 VGPR):   VGPR r -> row r (lanes 0-15) / row r+8 (16-31)
template <int DOUT, bool RELU>
__global__ __launch_bounds__(32) void sage_gemm_wmma(
    const float* __restrict__ h,       // [N_NODES, 128]
    const float* __restrict__ agg,     // [N_NODES, 128]
    const float* __restrict__ invdeg,  // [N_NODES]
    const float* __restrict__ Wself,   // [128, DOUT] row-major
    const float* __restrict__ Wneigh,  // [128, DOUT] row-major
    const float* __restrict__ bias,    // [DOUT]
    float* __restrict__ out) {         // [N_NODES, DOUT]
  const int tile_m = blockIdx.x;        // 3125 tiles of 16 rows (exact)
  const int tile_n = blockIdx.y;        // DOUT/16 tiles of 16 cols (exact)
  const int lane   = threadIdx.x;       // 0..31, EXEC all ones
  const int mr     = lane & 15;
  const int klo    = (lane >> 4) * 2;   // 0 for lanes 0-15, 2 for 16-31
  const int row    = tile_m * 16 + mr;
  const int col    = tile_n * 16 + mr;

  const float idg = invdeg[row];
  const float* __restrict__ hrow = h   + (size_t)row * FEATS;
  const float* __restrict__ arow = agg + (size_t)row * FEATS;

  v8f acc = {};
#pragma unroll 4
  for (int k = 0; k < FEATS; k += 4) {
    v2f a, b;
    // self path: h @ Wself
    a.x = hrow[k + klo];
    a.y = hrow[k + klo + 1];
    b.x = Wself[(size_t)(k + klo) * DOUT + col];
    b.y = Wself[(size_t)(k + klo + 1) * DOUT + col];
    acc = __builtin_amdgcn_wmma_f32_16x16x4_f32(false, a, false, b,
                                                (short)0, acc, false, false);
    // neighbor path: (agg * invdeg) @ Wneigh   (invdeg folded into A-load)
    a.x = arow[k + klo] * idg;
    a.y = arow[k + klo + 1] * idg;
    b.x = Wneigh[(size_t)(k + klo) * DOUT + col];
    b.y = Wneigh[(size_t)(k + klo + 1) * DOUT + col];
    acc = __builtin_amdgcn_wmma_f32_16x16x4_f32(false, a, false, b,
                                                (short)0, acc, false, false);
  }

  const float bcol  = bias[col];
  const int   mbase = tile_m * 16 + ((lane >> 4) << 3);  // +0 or +8 row half
  const int   ncol  = tile_n * 16 + mr;
#pragma unroll
  for (int r = 0; r < 8; ++r) {
    float v = acc[r] + bcol;
    if (RELU) v = fmaxf(v, 0.0f);
    out[(size_t)(mbase + r) * DOUT + ncol] = v;
  }
}

// ----------------------------------------------------------------- dispatch
static void run_layer_128(const float* h_in, const int* src, const int* dst,
                          const float* invdeg, float* agg,
                          const float* Wself, const float* Wneigh,
                          const float* bias, float* h_out, hipStream_t stream) {
  sage_fill_zero<<<(N_NODES * FEATS + 255) / 256, 256, 0, stream>>>(agg, N_NODES * FEATS);
  sage_aggregate<<<(N_EDGES * 32) / 256, 256, 0, stream>>>(h_in, src, dst, agg);
  sage_gemm_wmma<128, true><<<dim3(N_NODES / 16, 128 / 16), 32, 0, stream>>>(
      h_in, agg, invdeg, Wself, Wneigh, bias, h_out);
}

extern "C" void kernel_launch(void* const* d_in, const int* in_sizes, int n_in,
                              void* d_out, int out_size, void* d_ws, size_t ws_size,
                              hipStream_t stream) {
  (void)in_sizes; (void)n_in; (void)out_size; (void)ws_size;

  const float* x       = (const float*)d_in[0];
  const int*   src     = (const int*)d_in[1];
  const int*   dst     = (const int*)d_in[2];
  const float* Wself0  = (const float*)d_in[3];
  const float* Wneigh0 = (const float*)d_in[4];
  const float* b0      = (const float*)d_in[5];
  const float* Wself1  = (const float*)d_in[6];
  const float* Wneigh1 = (const float*)d_in[7];
  const float* b1      = (const float*)d_in[8];
  const float* Wself2  = (const float*)d_in[9];
  const float* Wneigh2 = (const float*)d_in[10];
  const float* b2      = (const float*)d_in[11];

  float* ws     = (float*)d_ws;
  float* invdeg = ws;                                  // N_NODES floats
  float* agg    = invdeg + N_NODES;                    // N_NODES*128
  float* h1     = agg + (size_t)N_NODES * FEATS;       // N_NODES*128
  float* h2     = h1  + (size_t)N_NODES * FEATS;       // N_NODES*128
  float* out    = (float*)d_out;                       // N_NODES*64

  // shared inverse degree (clamped to 1)
  sage_fill_zero<<<(N_NODES + 255) / 256, 256, 0, stream>>>(invdeg, N_NODES);
  sage_degree<<<(N_EDGES + 255) / 256, 256, 0, stream>>>(dst, invdeg);
  sage_invdeg<<<(N_NODES + 255) / 256, 256, 0, stream>>>(invdeg);

  // layer 0: x -> h1 (relu), layer 1: h1 -> h2 (relu)
  run_layer_128(x,  src, dst, invdeg, agg, Wself0, Wneigh0, b0, h1, stream);
  run_layer_128(h1, src, dst, invdeg, agg, Wself1, Wneigh1, b1, h2, stream);

  // layer 2: h2 -> out (64 classes, no relu)
  sage_fill_zero<<<(N_NODES * FEATS + 255) / 256, 256, 0, stream>>>(agg, N_NODES * FEATS);
  sage_aggregate<<<(N_EDGES * 32) / 256, 256, 0, stream>>>(h2, src, dst, agg);
  sage_gemm_wmma<64, false><<<dim3(N_NODES / 16, 64 / 16), 32, 0, stream>>>(
      h2, agg, invdeg, Wself2, Wneigh2, b2, out);
}